// InstDecoder_69286412419670
// MI455X (gfx1250) — compile-verified
//
#include <hip/hip_runtime.h>

typedef float v2f __attribute__((ext_vector_type(2)));
typedef float v8f __attribute__((ext_vector_type(8)));

#define B_DIM 2
#define C_DIM 64
#define KD    64
#define M_DIM (8*256*256)      /* 524288 */
#define NUM_MASKS 100
#define NSEG 101               /* labels 0..100 */
#define NPAD 112               /* 7 tiles of 16 rows */
#define MT_CNT (M_DIM/16)      /* 32768 m-tiles per image */

/* ---- ws layout (floats) ----
   sums   : [B][101][64]  @ 0          (12928)
   counts : [B][101]      @ 12928      (202)
   instF  : [B][112][64]  @ 13184      (14336)   (128B aligned)
   predk  : [B][112][64]  @ 27520      (14336)
*/
#define WS_SUMS   0
#define WS_COUNTS 12928
#define WS_INSTF  13184
#define WS_PREDK  27520
#define WS_ZERO_N 13130        /* sums + counts */

__global__ void zero_ws_kernel(float* __restrict__ p, int n) {
    int i = blockIdx.x * blockDim.x + threadIdx.x;
    if (i < n) p[i] = 0.0f;
}

/* Stage 1: per-block LDS segment accumulation, then global atomic flush. */
#define PIX_PER_BLOCK 2048
__global__ void __launch_bounds__(256)
seg_accum_kernel(const float* __restrict__ feat, const int* __restrict__ labels,
                 float* __restrict__ sums, float* __restrict__ counts) {
    __shared__ float lsum[NSEG * C_DIM];   /* 25.8 KB */
    __shared__ float lcnt[NSEG];
    const int b = blockIdx.y;
    const long base = (long)blockIdx.x * PIX_PER_BLOCK;

    for (int i = threadIdx.x; i < NSEG * C_DIM; i += blockDim.x) lsum[i] = 0.0f;
    for (int i = threadIdx.x; i < NSEG;        i += blockDim.x) lcnt[i] = 0.0f;
    __syncthreads();

    for (int p = threadIdx.x; p < PIX_PER_BLOCK; p += blockDim.x) {
        const long m = base + p;
        const int lab = labels[(long)b * M_DIM + m];
        if (lab > 0) {
            atomicAdd(&lcnt[lab], 1.0f);
            const float* f = feat + (long)b * C_DIM * M_DIM + m;
            #pragma unroll 8
            for (int c = 0; c < C_DIM; ++c)
                atomicAdd(&lsum[lab * C_DIM + c], f[(long)c * M_DIM]);
        }
    }
    __syncthreads();

    for (int i = threadIdx.x; i < NSEG * C_DIM; i += blockDim.x) {
        const float v = lsum[i];
        if (v != 0.0f) atomicAdd(&sums[(long)b * NSEG * C_DIM + i], v);
    }
    for (int i = threadIdx.x; i < NSEG; i += blockDim.x) {
        const float v = lcnt[i];
        if (v != 0.0f) atomicAdd(&counts[(long)b * NSEG + i], v);
    }
}

/* Stage 1.5: masked mean, write padded [B][112][64] (rows >= 100 are zero). */
__global__ void seg_mean_kernel(const float* __restrict__ sums,
                                const float* __restrict__ counts,
                                float* __restrict__ instF) {
    int idx = blockIdx.x * blockDim.x + threadIdx.x;   /* B*112*64 = 14336 */
    if (idx >= B_DIM * NPAD * C_DIM) return;
    const int c   = idx & 63;
    const int row = (idx >> 6) % NPAD;
    const int b   = idx / (NPAD * C_DIM);
    float v = 0.0f;
    if (row < NUM_MASKS) {
        const float cnt = counts[b * NSEG + row + 1];
        v = sums[((long)b * NSEG + row + 1) * C_DIM + c] / fmaxf(cnt, 1.0f);
    }
    instF[idx] = v;
}

/* Stage 2: predk[b] = instF[b](112x64) @ Wk(64x64) + bk, one 16x16 tile per wave.
   56 waves total (2 images * 7 row-tiles * 4 col-tiles). */
__global__ void __launch_bounds__(256)
predk_gemm_kernel(const float* __restrict__ instF, const float* __restrict__ Wk,
                  const float* __restrict__ bk, float* __restrict__ predk) {
    const int wid  = (blockIdx.x * blockDim.x + threadIdx.x) >> 5;
    const int lane = threadIdx.x & 31;
    const int b    = wid / 28;
    const int rem  = wid % 28;
    const int nt   = rem >> 2;
    const int jt   = rem & 3;
    const int lrow = lane & 15;       /* A row / B-D col */
    const int half = lane >> 4;

    v8f acc = {};
    const float* A = instF + ((long)b * NPAD + nt * 16 + lrow) * KD;
    #pragma unroll
    for (int kk = 0; kk < 16; ++kk) {
        const int kb = 4 * kk + 2 * half;
        const v2f a = *(const v2f*)(A + kb);
        v2f bb;
        bb.x = Wk[(long)kb       * KD + jt * 16 + lrow];
        bb.y = Wk[(long)(kb + 1) * KD + jt * 16 + lrow];
        acc = __builtin_amdgcn_wmma_f32_16x16x4_f32(false, a, false, bb,
                                                    (short)0, acc, false, false);
    }
    const float bias = bk[jt * 16 + lrow];
    #pragma unroll
    for (int r = 0; r < 8; ++r) {
        const int n = nt * 16 + r + 8 * half;
        predk[((long)b * NPAD + n) * KD + jt * 16 + lrow] = acc[r] + bias;
    }
}

/* Stage 3: pred_masks[b] = predk[b](112x64) @ mask_features[b](64xM).
   One wave per (b, 16-wide m-tile); B fragment reused across all 7 n-tiles
   so mask_features streams from HBM exactly once. */
__global__ void __launch_bounds__(256)
mask_gemm_kernel(const float* __restrict__ predk, const float* __restrict__ mf,
                 float* __restrict__ out) {
    const int wid  = (blockIdx.x * blockDim.x + threadIdx.x) >> 5;
    const int lane = threadIdx.x & 31;
    const int b    = wid >> 15;        /* / 32768 */
    const int mt   = wid & (MT_CNT - 1);
    const long m0  = (long)mt * 16;
    const int lrow = lane & 15;
    const int half = lane >> 4;

    v8f acc[7];
    #pragma unroll
    for (int i = 0; i < 7; ++i) acc[i] = (v8f){};

    const float* mfb = mf + (long)b * KD * M_DIM + m0 + lrow;
    const float* pkb = predk + (long)b * NPAD * KD;

    for (int kk = 0; kk < 16; ++kk) {
        const int kb = 4 * kk + 2 * half;
        v2f bb;
        bb.x = mfb[(long)kb       * M_DIM];
        bb.y = mfb[(long)(kb + 1) * M_DIM];
        #pragma unroll
        for (int nt = 0; nt < 7; ++nt) {
            const v2f a = *(const v2f*)(pkb + (nt * 16 + lrow) * KD + kb);
            acc[nt] = __builtin_amdgcn_wmma_f32_16x16x4_f32(false, a, false, bb,
                                                            (short)0, acc[nt],
                                                            false, false);
        }
    }

    #pragma unroll
    for (int nt = 0; nt < 7; ++nt) {
        #pragma unroll
        for (int r = 0; r < 8; ++r) {
            const int n = nt * 16 + r + 8 * half;
            if (n < NUM_MASKS)
                out[((long)b * NUM_MASKS + n) * M_DIM + m0 + lrow] = acc[nt][r];
        }
    }
}

extern "C" void kernel_launch(void* const* d_in, const int* in_sizes, int n_in,
                              void* d_out, int out_size, void* d_ws, size_t ws_size,
                              hipStream_t stream) {
    (void)in_sizes; (void)n_in; (void)out_size; (void)ws_size;
    const float* features      = (const float*)d_in[0];
    const float* mask_features = (const float*)d_in[1];
    const float* Wk            = (const float*)d_in[2];
    const float* bk            = (const float*)d_in[3];
    const int*   init_masks    = (const int*)d_in[4];
    float* out = (float*)d_out;

    float* ws     = (float*)d_ws;
    float* sums   = ws + WS_SUMS;
    float* counts = ws + WS_COUNTS;
    float* instF  = ws + WS_INSTF;
    float* predk  = ws + WS_PREDK;

    /* zero the accumulators (harness does not re-zero between replays) */
    zero_ws_kernel<<<(WS_ZERO_N + 255) / 256, 256, 0, stream>>>(ws, WS_ZERO_N);

    /* stage 1: segment sums + counts */
    seg_accum_kernel<<<dim3(M_DIM / PIX_PER_BLOCK, B_DIM), 256, 0, stream>>>(
        features, init_masks, sums, counts);

    /* stage 1.5: masked means -> padded instF */
    seg_mean_kernel<<<(B_DIM * NPAD * C_DIM + 255) / 256, 256, 0, stream>>>(
        sums, counts, instF);

    /* stage 2: instF @ Wk + bk  (56 waves) */
    predk_gemm_kernel<<<7, 256, 0, stream>>>(instF, Wk, bk, predk);

    /* stage 3: predk @ mask_features  (65536 waves, 8 per block) */
    mask_gemm_kernel<<<(B_DIM * MT_CNT) / 8, 256, 0, stream>>>(
        predk, mask_features, out);
}